// MultiScaleRNN_69028714382012
// MI455X (gfx1250) — compile-verified
//
#include <hip/hip_runtime.h>
#include <hip/hip_bf16.h>

typedef __attribute__((ext_vector_type(2))) float v2f;
typedef __attribute__((ext_vector_type(8))) float v8f;

#define T_LEN   65536
#define D_IN    128
#define H_DIM   128
#define G3      384            // 3*H
#define N_HEADS 8
#define OUT_W   (N_HEADS * H_DIM)   // 1024

// ---------------------------------------------------------------------------
// Phase 1: ig[head][t][g] = sum_d x[t][d] * Wih[head][g][d] + b[head][g]
// GEMM C[T, 3072]:  A = x [T,128] row-major, B(n,k) = Wih_flat[n*128 + k].
// One wave computes a 16(M) x 64(N) strip using V_WMMA_F32_16X16X4_F32,
// 4 independent accumulator chains reuse each A fragment.
// ---------------------------------------------------------------------------
__global__ __launch_bounds__(256)
void gru_igates_wmma(const float* __restrict__ x,
                     const float* __restrict__ wih,
                     const float* __restrict__ bias,
                     float* __restrict__ ig) {
  const int wave = threadIdx.x >> 5;
  const int lane = threadIdx.x & 31;
  const int half = lane >> 4;       // 0: lanes 0-15, 1: lanes 16-31
  const int lm   = lane & 15;

  const long w  = (long)blockIdx.x * 8 + wave;   // global wave id
  const int gn  = (int)(w % 48);                 // 48 groups of 64 cols
  const int gm  = (int)(w / 48);                 // 4096 M tiles
  const int m0  = gm * 16;
  const int n0  = gn * 64;

  v8f c0 = {}, c1 = {}, c2 = {}, c3 = {};

  // A fragment base: lane holds x[m0+lm][kk + 2*half + {0,1}]
  const float* arow = x + (size_t)(m0 + lm) * D_IN + 2 * half;
  // B fragment base for sub-tile j: B(n0+16j+lm, kk + 2*half + {0,1})
  const float* brow = wih + (size_t)(n0 + lm) * D_IN + 2 * half;

  #pragma unroll 4
  for (int kk = 0; kk < D_IN; kk += 4) {
    float2 af = *(const float2*)(arow + kk);
    v2f a; a.x = af.x; a.y = af.y;
    v2f b; float2 bf;

    bf = *(const float2*)(brow + kk);                 b.x = bf.x; b.y = bf.y;
    c0 = __builtin_amdgcn_wmma_f32_16x16x4_f32(false, a, false, b, (short)0, c0, false, false);
    bf = *(const float2*)(brow + 16 * D_IN + kk);     b.x = bf.x; b.y = bf.y;
    c1 = __builtin_amdgcn_wmma_f32_16x16x4_f32(false, a, false, b, (short)0, c1, false, false);
    bf = *(const float2*)(brow + 32 * D_IN + kk);     b.x = bf.x; b.y = bf.y;
    c2 = __builtin_amdgcn_wmma_f32_16x16x4_f32(false, a, false, b, (short)0, c2, false, false);
    bf = *(const float2*)(brow + 48 * D_IN + kk);     b.x = bf.x; b.y = bf.y;
    c3 = __builtin_amdgcn_wmma_f32_16x16x4_f32(false, a, false, b, (short)0, c3, false, false);
  }

  // Bias per column (same for all rows of the tile).
  const float bias0 = bias[n0 + lm];
  const float bias1 = bias[n0 + 16 + lm];
  const float bias2 = bias[n0 + 32 + lm];
  const float bias3 = bias[n0 + 48 + lm];

  // Output: ig laid out [head][t][g];  n = head*384 + g, 64 | 384 so one head.
  const int headN = n0 / G3;
  const int gb    = n0 % G3;
  float* base = ig + (size_t)headN * T_LEN * G3;

  #pragma unroll
  for (int r = 0; r < 8; ++r) {
    const int m = m0 + r + 8 * half;           // C layout: M = r + 8*(lane>=16)
    float* rowp = base + (size_t)m * G3 + gb + lm;
    rowp[0]  = c0[r] + bias0;
    rowp[16] = c1[r] + bias1;
    rowp[32] = c2[r] + bias2;
    rowp[48] = c3[r] + bias3;
  }
}

// ---------------------------------------------------------------------------
// Phase 2: latency-optimized GRU scan. One block per independent chain
// (255 chains: head i contributes 2^i chains of length T>>i).
// Thread g (0..383) holds Whh[head][g][0:128] in registers; h lives in LDS.
// ig row for t+1 is prefetched while step t computes.
// ---------------------------------------------------------------------------
__global__ __launch_bounds__(384, 1)
void gru_scan(const float* __restrict__ ig_all,
              const float* __restrict__ whh,
              const float* __restrict__ bn,
              float* __restrict__ out) {
  __shared__ float h_lds[H_DIM];
  __shared__ float hg_lds[G3];
  __shared__ float igl[G3];

  const int g     = threadIdx.x;            // 0..383
  const int node  = blockIdx.x + 1;         // 1..255 (heap index)
  const int head  = 31 - __clz(node);
  const int chain = node - (1 << head);
  const int Tc    = T_LEN >> head;

  // Recurrent weights -> registers (fully unrolled, static indices).
  float w[H_DIM];
  const float* wp = whh + ((size_t)head * G3 + g) * H_DIM;
  #pragma unroll
  for (int k = 0; k < H_DIM; ++k) w[k] = wp[k];

  float bnv = 0.0f;
  if (g < H_DIM) bnv = bn[head * H_DIM + g];

  const float* igp  = ig_all + ((size_t)head * T_LEN + (size_t)chain * Tc) * G3;
  float*       outp = out + (size_t)chain * Tc * OUT_W + head * H_DIM;

  if (g < H_DIM) h_lds[g] = 0.0f;
  float ig_cur = igp[g];                    // ig[0][g]
  __syncthreads();

  for (int t = 0; t < Tc; ++t) {
    // hg[g] = sum_k Whh[g][k] * h[k]   (broadcast LDS reads)
    float acc = 0.0f;
    #pragma unroll
    for (int k = 0; k < H_DIM; ++k) acc = fmaf(w[k], h_lds[k], acc);
    hg_lds[g] = acc;
    igl[g]    = ig_cur;

    // Prefetch next ig row; latency hides under barrier + gate math + matvec.
    const int tn = (t + 1 < Tc) ? (t + 1) : (Tc - 1);
    ig_cur = igp[(size_t)tn * G3 + g];

    __syncthreads();

    if (g < H_DIM) {
      const float ir = igl[g], iz = igl[g + 128], in_ = igl[g + 256];
      const float hr = hg_lds[g], hz = hg_lds[g + 128], hn = hg_lds[g + 256];
      const float r  = 1.0f / (1.0f + __expf(-(ir + hr)));
      const float z  = 1.0f / (1.0f + __expf(-(iz + hz)));
      const float pn = in_ + r * (hn + bnv);
      const float n  = 2.0f / (1.0f + __expf(-2.0f * pn)) - 1.0f;   // tanh(pn)
      const float ho = h_lds[g];
      const float hnew = n + z * (ho - n);
      h_lds[g] = hnew;
      outp[(size_t)t * OUT_W + g] = hnew;   // fire-and-forget store
    }
    __syncthreads();
  }
}

// ---------------------------------------------------------------------------
extern "C" void kernel_launch(void* const* d_in, const int* in_sizes, int n_in,
                              void* d_out, int out_size, void* d_ws, size_t ws_size,
                              hipStream_t stream) {
  const float* x   = (const float*)d_in[0];   // [T, 128]
  const float* wih = (const float*)d_in[1];   // [8, 384, 128]
  const float* whh = (const float*)d_in[2];   // [8, 384, 128]
  const float* b   = (const float*)d_in[3];   // [8, 384]
  const float* bn  = (const float*)d_in[4];   // [8, 128]
  float* out = (float*)d_out;                 // [T, 1024]
  float* ig  = (float*)d_ws;                  // [8][T][384] f32 = 768 MB

  // Phase 1: 4096 M-tiles x 48 N-groups = 196608 waves, 8 waves per block.
  const unsigned grid1 = (unsigned)(((long)(T_LEN / 16) * 48) / 8);  // 24576
  gru_igates_wmma<<<dim3(grid1), dim3(256), 0, stream>>>(x, wih, b, ig);

  // Phase 2: one block per chain (sum_{i<8} 2^i = 255 chains).
  gru_scan<<<dim3(255), dim3(384), 0, stream>>>(ig, whh, bn, out);
}